// RotaryPosEmbLayer_17446157156699
// MI455X (gfx1250) — compile-verified
//
#include <hip/hip_runtime.h>
#include <cstdint>

#define BLOCK 256
#define TCHUNK 16

// Types matching clang's declaration of the gfx1250 async-to-LDS builtin:
//   param0: AS1 pointer to 16B int vector, param1: AS3 pointer to same.
typedef int v4i __attribute__((vector_size(4 * sizeof(int))));
typedef __attribute__((address_space(1))) v4i gv4i_t;
typedef __attribute__((address_space(3))) v4i lv4i_t;

// Rotary-embedding scan, solved in closed form: R_t = Rot((t+1)*theta_h) @ R_0.
// Memory-bound fp32 streaming (~256 MB @ 23.3 TB/s ~= 11 us floor).
__global__ __launch_bounds__(BLOCK) void rope_scan_kernel(
    const float* __restrict__ x,     // [T, 2*half]
    const float* __restrict__ R0,    // [half, 2, 2]
    float* __restrict__ y,           // [T, 2*half]      (inputs_rot)
    float* __restrict__ Rall,        // [T, half, 2, 2]  (R_all)
    int T, int half)
{
    __shared__ float ldsR[BLOCK * 4];

    const int tid = threadIdx.x;
    const int h   = blockIdx.x * BLOCK + tid;
    const int t0  = blockIdx.y * TCHUNK;
    const int D   = 2 * half;

    // ---- Stage this block's slice of R_tminus1 into LDS (gfx1250 async path) ----
    if (h < half) {
        const float* gsrc = R0 + (size_t)h * 4;   // 16B per thread, lane-contiguous
        float*       ldst = &ldsR[tid * 4];
#if __has_builtin(__builtin_amdgcn_global_load_async_to_lds_b128)
        __builtin_amdgcn_global_load_async_to_lds_b128(
            (gv4i_t*)(uintptr_t)gsrc,
            (lv4i_t*)(uintptr_t)ldst,
            /*offset=*/0, /*cpol=*/0);
  #if __has_builtin(__builtin_amdgcn_s_wait_asynccnt)
        __builtin_amdgcn_s_wait_asynccnt(0);
  #else
        asm volatile("s_wait_asynccnt 0" ::: "memory");
  #endif
#else
        // Fallback: synchronous copy through VGPRs.
        *(float4*)ldst = *(const float4*)gsrc;
#endif
    }
    __syncthreads();

    if (h >= half) return;

    // R_prev for this pair (ds_load_b128)
    const float4 pv = *(const float4*)&ldsR[tid * 4];
    const float p00 = pv.x, p01 = pv.y, p10 = pv.z, p11 = pv.w;

    // theta_h = exp(2h * (-ln(10000)/d_model)) = exp(h * (-ln(10000)/half))
    const float theta = __expf((float)h * (-9.210340371976184f / (float)half));

    // Step rotation (angle-addition recurrence), computed once per thread.
    float ct, st;
    __sincosf(theta, &st, &ct);

    // Base angle for this chunk: (t0+1)*theta. Accurate sincos once per thread.
    float S, C;
    sincosf((float)(t0 + 1) * theta, &S, &C);

    const float* xp = x    + (size_t)t0 * D + 2 * h;
    float*       yp = y    + (size_t)t0 * D + 2 * h;
    float*       rp = Rall + (size_t)t0 * (size_t)(half * 4) + 4 * h;

#pragma unroll
    for (int k = 0; k < TCHUNK; ++k) {
        if (t0 + k < T) {
            // R_t = Rot(angle) @ R_prev
            const float r00 = C * p00 - S * p10;
            const float r01 = C * p01 - S * p11;
            const float r10 = S * p00 + C * p10;
            const float r11 = S * p01 + C * p11;

            // x_rot = R_t @ x_pair  (b64 load/store, lane-contiguous)
            const float2 xv = *(const float2*)xp;
            float2 yv;
            yv.x = r00 * xv.x + r01 * xv.y;
            yv.y = r10 * xv.x + r11 * xv.y;
            *(float2*)yp = yv;

            // R_all[t][h] (b128 store, lane-contiguous)
            *(float4*)rp = make_float4(r00, r01, r10, r11);
        }
        xp += D; yp += D; rp += (size_t)half * 4;

        // advance (C,S) by theta: 4 FMAs instead of another sincos
        const float Cn = C * ct - S * st;
        const float Sn = S * ct + C * st;
        C = Cn; S = Sn;
    }
}

extern "C" void kernel_launch(void* const* d_in, const int* in_sizes, int n_in,
                              void* d_out, int out_size, void* d_ws, size_t ws_size,
                              hipStream_t stream) {
    (void)n_in; (void)d_ws; (void)ws_size; (void)out_size;

    const float* x  = (const float*)d_in[0];   // [T, D] fp32
    const float* R0 = (const float*)d_in[1];   // [half, 2, 2] fp32

    const int half = in_sizes[1] / 4;          // half*2*2 elements
    const int D    = 2 * half;
    const int T    = in_sizes[0] / D;

    float* y    = (float*)d_out;               // inputs_rot: T*D floats
    float* Rall = y + (size_t)T * D;           // R_all: T*half*4 floats

    dim3 block(BLOCK, 1, 1);
    dim3 grid((half + BLOCK - 1) / BLOCK, (T + TCHUNK - 1) / TCHUNK, 1);
    rope_scan_kernel<<<grid, block, 0, stream>>>(x, R0, y, Rall, T, half);
}